// _ConditionalInstanceNorm_48335561949473
// MI455X (gfx1250) — compile-verified
//
#include <hip/hip_runtime.h>

// Conditional Instance Norm, CDNA5 (gfx1250).
// x: [16,64,256,256] f32; styles: [16] i32; gamma/beta: [4,64] f32.
// One workgroup per (b,c) plane. Plane (256 KB) is staged in LDS (320 KB/WGP
// on CDNA5) via GLOBAL_LOAD_ASYNC_TO_LDS_B128, reduced with
// V_WMMA_F32_16X16X4_F32 (B = ones -> row sums), then normalized out of LDS
// with nontemporal b128 stores. HBM traffic = 1 read + 1 write (the floor).

typedef float v2f __attribute__((ext_vector_type(2)));
typedef float v4f __attribute__((ext_vector_type(4)));
typedef float v8f __attribute__((ext_vector_type(8)));
typedef int   v4i __attribute__((ext_vector_type(4)));

#define B_DIM   16
#define C_DIM   64
#define HW      65536           // 256*256
#define THREADS 256             // 8 wave32
#define EPS     1e-5f

// Builtin expects: (v4i addrspace(1)*, v4i addrspace(3)*, int, int)
typedef __attribute__((address_space(1))) v4i* g4p;
typedef __attribute__((address_space(3))) v4i* l4p;

__global__ __launch_bounds__(THREADS)
void cin_kernel(const float* __restrict__ x,
                const int*   __restrict__ styles,
                const float* __restrict__ gamma,
                const float* __restrict__ beta,
                float*       __restrict__ out) {
  extern __shared__ float tile[];          // HW floats = 256 KB
  __shared__ float s_sum[8];
  __shared__ float s_sq[8];
  __shared__ float s_bc[2];

  const int bc  = blockIdx.x;              // 0..1023
  const int b   = bc >> 6;                 // / C_DIM
  const int c   = bc & (C_DIM - 1);
  const int tid = threadIdx.x;
  const size_t base = (size_t)bc * (size_t)HW;
  const float* xp = x + base;
  float*       op = out + base;

  // ---------------- Stage 1: plane global -> LDS (async on CDNA5) ----------
  {
    const v4f* xp4 = (const v4f*)xp;       // 16384 float4 per plane
#if __has_builtin(__builtin_amdgcn_global_load_async_to_lds_b128)
    #pragma unroll
    for (int i = 0; i < 64; ++i) {
      const int idx = i * THREADS + tid;
      __builtin_amdgcn_global_load_async_to_lds_b128(
          (g4p)(xp4 + idx),
          (l4p)(tile + (idx << 2)),
          0, 0);
    }
#if __has_builtin(__builtin_amdgcn_s_wait_asynccnt)
    __builtin_amdgcn_s_wait_asynccnt(0);
#else
    asm volatile("s_wait_asynccnt 0" ::: "memory");
#endif
#else
    v4f* t4 = (v4f*)tile;
    #pragma unroll
    for (int i = 0; i < 64; ++i) {
      const int idx = i * THREADS + tid;
      t4[idx] = xp4[idx];
    }
#endif
  }
  __syncthreads();

  // ---------------- Stage 2: sum / sumsq via V_WMMA_F32_16X16X4_F32 --------
  // B = all-ones: D[m][n] = rowsum_m for every n, so every A slot contributes
  // exactly once to column 0 of D -> data placement into A slots is free.
  const int wave = tid >> 5;
  const int lane = tid & 31;
  v8f accS = {};
  v8f accQ = {};
  const v2f onesB = {1.0f, 1.0f};
  const v2f* wbase = (const v2f*)tile + wave * 4096;   // 8192 floats per wave
  #pragma unroll 4
  for (int it = 0; it < 128; ++it) {
    v2f a = wbase[it * 32 + lane];                      // ds_load_b64, no bank conflicts
    v2f q = a * a;
    accS = __builtin_amdgcn_wmma_f32_16x16x4_f32(
        false, a, false, onesB, (short)0, accS, false, false);
    accQ = __builtin_amdgcn_wmma_f32_16x16x4_f32(
        false, q, false, onesB, (short)0, accQ, false, false);
  }
  float s8 = accS[0] + accS[1] + accS[2] + accS[3] +
             accS[4] + accS[5] + accS[6] + accS[7];
  float q8 = accQ[0] + accQ[1] + accQ[2] + accQ[3] +
             accQ[4] + accQ[5] + accQ[6] + accQ[7];
  // Column 0 of D: lanes 0 (M=0..7 in regs 0..7) and 16 (M=8..15).
  float wsum = __shfl(s8, 0, 32) + __shfl(s8, 16, 32);
  float wsq  = __shfl(q8, 0, 32) + __shfl(q8, 16, 32);
  if (lane == 0) { s_sum[wave] = wsum; s_sq[wave] = wsq; }
  __syncthreads();

  if (tid == 0) {
    float S = 0.f, Q = 0.f;
    #pragma unroll
    for (int w = 0; w < 8; ++w) { S += s_sum[w]; Q += s_sq[w]; }
    const float mean = S * (1.0f / (float)HW);
    const float var  = Q * (1.0f / (float)HW) - mean * mean;
    s_bc[0] = mean;
    s_bc[1] = rsqrtf(var + EPS);
  }
  __syncthreads();

  const float mean = s_bc[0];
  const float inv  = s_bc[1];
  const int   st   = styles[b];
  const float g    = gamma[st * C_DIM + c];
  const float be   = beta[st * C_DIM + c];
  const float scale = inv * g;
  const float shift = be - mean * scale;

  // ---------------- Stage 3: normalize LDS -> global (NT b128 stores) ------
  const v4f* t4 = (const v4f*)tile;
  v4f*       o4 = (v4f*)op;
  #pragma unroll
  for (int i = 0; i < 64; ++i) {
    const int idx = i * THREADS + tid;
    v4f v = t4[idx];
    v4f r = v * scale + shift;               // v_fma per component
    __builtin_nontemporal_store(r, o4 + idx);
  }
}

extern "C" void kernel_launch(void* const* d_in, const int* in_sizes, int n_in,
                              void* d_out, int out_size, void* d_ws, size_t ws_size,
                              hipStream_t stream) {
  const float* x      = (const float*)d_in[0];
  const int*   styles = (const int*)  d_in[1];
  const float* gamma  = (const float*)d_in[2];
  const float* beta   = (const float*)d_in[3];
  float*       out    = (float*)d_out;

  const size_t lds_bytes = (size_t)HW * sizeof(float);   // 256 KB < 320 KB/WGP
  // Raise the dynamic-LDS cap (idempotent; executes immediately, capture-safe).
  (void)hipFuncSetAttribute((const void*)cin_kernel,
                            hipFuncAttributeMaxDynamicSharedMemorySize,
                            (int)lds_bytes);

  cin_kernel<<<B_DIM * C_DIM, THREADS, lds_bytes, stream>>>(x, styles, gamma,
                                                            beta, out);
}